// PermutationCrossEntropy_28681791602913
// MI455X (gfx1250) — compile-verified
//
#include <hip/hip_runtime.h>
#include <math.h>

typedef __attribute__((ext_vector_type(2))) float v2f;
typedef __attribute__((ext_vector_type(4))) float v4f;
typedef __attribute__((ext_vector_type(8))) float v8f;

#define NPERM 4
#define NCLS  512

// itertools.permutations(range(4)) order, packed 2 bits per slot:
// PERMPACK[n] bits [2p+1:2p] = pi_n(p). Entries 24..31 are padding (value
// irrelevant: those WMMA columns carry C=+INF and are discarded by the min).
__device__ __constant__ unsigned int PERMPACK[32] = {
    0xE4, 0xB4, 0xD8, 0x78, 0x9C, 0x6C,
    0xE1, 0xB1, 0xC9, 0x39, 0x8D, 0x2D,
    0xD2, 0x72, 0xC6, 0x36, 0x4E, 0x1E,
    0x93, 0x63, 0x87, 0x27, 0x4B, 0x1B,
    0xFF, 0xFF, 0xFF, 0xFF, 0xFF, 0xFF, 0xFF, 0xFF};

// Selection matrix S[k][n]: 1 if permutation n maps slot p=k>>2 to target q=k&3.
__device__ __forceinline__ float permsel(int k, int n) {
  int p = k >> 2, q = k & 3;
  return (((PERMPACK[n] >> (2 * p)) & 3u) == (unsigned)q) ? 1.0f : 0.0f;
}

__global__ __launch_bounds__(512) void perm_ce_kernel(
    const float* __restrict__ preds, const int* __restrict__ targets,
    float* __restrict__ out) {
  // 16 waves; wave w reduces batch b = blockIdx*16 + w (4 rows of 512 logits).
  __shared__ float G[16][17];  // padded: stride-17 avoids bank conflicts on A reads

  const int lane = threadIdx.x & 31;
  const int wave = threadIdx.x >> 5;
  const int b    = blockIdx.x * 16 + wave;
  const size_t bBase = (size_t)b * NPERM * NCLS;

  for (int p = 0; p < NPERM; ++p) {
    const float* row = preds + bBase + (size_t)p * NCLS;

    // Gather target logits FIRST (lanes 0..3), so NT streaming can't hurt us.
    float tval = 0.0f;
    if (lane < 4) {
      int tq = targets[b * 4 + lane];
      tval = row[tq];
    }

    // Stream the row: 4 x b128 coalesced NT loads per lane (16 floats/lane).
    v4f v[4];
#pragma unroll
    for (int t = 0; t < 4; ++t) {
      const v4f* ptr = (const v4f*)(row + t * 128 + lane * 4);
      v[t] = __builtin_nontemporal_load(ptr);
    }

    // Row max (local 16 -> wave32 xor-shuffle reduce).
    float m = -INFINITY;
#pragma unroll
    for (int t = 0; t < 4; ++t)
      m = fmaxf(m, fmaxf(fmaxf(v[t].x, v[t].y), fmaxf(v[t].z, v[t].w)));
#pragma unroll
    for (int mask = 16; mask >= 1; mask >>= 1)
      m = fmaxf(m, __shfl_xor(m, mask, 32));

    // Sum of exp(x - max).
    float s = 0.0f;
#pragma unroll
    for (int t = 0; t < 4; ++t)
      s += __expf(v[t].x - m) + __expf(v[t].y - m) + __expf(v[t].z - m) +
           __expf(v[t].w - m);
#pragma unroll
    for (int mask = 16; mask >= 1; mask >>= 1)
      s += __shfl_xor(s, mask, 32);

    float lse = m + __logf(s);

    // CE term (positive): -logp[target] = lse - logit[target]. k = p*4+q.
    if (lane < 4) G[wave][p * 4 + lane] = lse - tval;
  }
  __syncthreads();

  // Wave 0: losses[16 batches x 24 perms] = G(16x16) @ S(16x24) via chained
  // V_WMMA_F32_16X16X4_F32 (4 K-steps), two N-tiles (perms 0..15, 16..23+pad).
  if (wave == 0) {
    const int n  = lane & 15;  // A: row M; B/D: column N (per 16-lane half)
    const int hi = lane >> 4;  // selects K pair {0,1} vs {2,3} for A/B frags

    v8f d1 = {};  // C = 0 for valid perm columns
    v8f d2;
    const float pad = (16 + n >= 24) ? INFINITY : 0.0f;  // kill pad columns
#pragma unroll
    for (int i = 0; i < 8; ++i) d2[i] = pad;

#pragma unroll
    for (int kk = 0; kk < 4; ++kk) {
      const int kb = kk * 4 + hi * 2;
      v2f a;  a.x  = G[n][kb];            a.y  = G[n][kb + 1];
      v2f b1; b1.x = permsel(kb, n);      b1.y = permsel(kb + 1, n);
      v2f b2; b2.x = permsel(kb, 16 + n); b2.y = permsel(kb + 1, 16 + n);
      d1 = __builtin_amdgcn_wmma_f32_16x16x4_f32(false, a, false, b1,
                                                 (short)0, d1, false, false);
      d2 = __builtin_amdgcn_wmma_f32_16x16x4_f32(false, a, false, b2,
                                                 (short)0, d2, false, false);
    }

    // Min over permutations: combine both N-tiles, then xor-min across the
    // 16-lane halves. Lane 0 holds rows M=0..7, lane 16 rows M=8..15.
    v8f d;
#pragma unroll
    for (int i = 0; i < 8; ++i) d[i] = fminf(d1[i], d2[i]);
#pragma unroll
    for (int mask = 1; mask <= 8; mask <<= 1) {
#pragma unroll
      for (int i = 0; i < 8; ++i) d[i] = fminf(d[i], __shfl_xor(d[i], mask, 32));
    }

    const int outBase = blockIdx.x * 16;
    if (lane == 0) {
#pragma unroll
      for (int i = 0; i < 8; ++i) out[outBase + i] = d[i];
    } else if (lane == 16) {
#pragma unroll
      for (int i = 0; i < 8; ++i) out[outBase + 8 + i] = d[i];
    }
  }
}

extern "C" void kernel_launch(void* const* d_in, const int* in_sizes, int n_in,
                              void* d_out, int out_size, void* d_ws,
                              size_t ws_size, hipStream_t stream) {
  const float* preds   = (const float*)d_in[0];
  const int*   targets = (const int*)d_in[1];
  float*       out     = (float*)d_out;

  const int B = in_sizes[1] / NPERM;  // targets is [B, 4]
  dim3 grid(B / 16), block(512);
  hipLaunchKernelGGL(perm_ce_kernel, grid, block, 0, stream, preds, targets,
                     out);
}